// CTCLayer_84971632984337
// MI455X (gfx1250) — compile-verified
//
#include <hip/hip_runtime.h>

// CTC batch cost for B=512, T=512, C=128, L=80 on gfx1250 (MI455X).
//
// log_softmax(log(y+eps))_c = log(y_c+eps) - log(sum_c (y_c+eps))
//   -> softmax needs only a channel-sum + log (no exp).
//
// Mapping: 1 workgroup (256 threads, 8 wave32) per batch row.
//   - time tiled by 16; 16x128 fp32 slab double-buffered in LDS
//   - next tile fetched with GLOBAL_LOAD_ASYNC_TO_LDS_B128 (ASYNCcnt),
//     overlapping the 16-step sequential scan; tile+2 warmed via prefetch
//   - 16 softmax denominators per tile via V_WMMA_F32_16X16X4_F32
//     (fp32-exact: A = 16x4 y-tile, B = ones), 32 accumulating WMMAs
//   - forward scan: thread s owns extended-label state s (S = 2L+1 = 161),
//     alpha in LDS, 2 barriers per time step

typedef __attribute__((ext_vector_type(2))) float v2f;
typedef __attribute__((ext_vector_type(8))) float v8f;
typedef __attribute__((ext_vector_type(4))) int   v4i;

typedef __attribute__((address_space(1))) v4i* gptr_v4i;  // global int4*
typedef __attribute__((address_space(3))) v4i* lptr_v4i;  // LDS int4*

#define B_    512
#define T_    512
#define C_    128
#define L_    80
#define S_    161          // 2*L+1
#define BLANK_ 127
#define NEG_  (-1.0e30f)
#define EPS_  (1.0e-7f)
#define TT    16           // time tile
#define NTILE (T_ / TT)    // 32

__device__ __forceinline__ void async_stage_tile(const float* gbase, float* lbase, int tid)
{
    // 16x128 fp32 slab = 8 KB = 512 int4; 256 threads x 2 b128 async loads.
    const char* g = (const char*)gbase + (size_t)tid * 16;
    char*       l = (char*)lbase       + (size_t)tid * 16;
    __builtin_amdgcn_global_load_async_to_lds_b128(
        (gptr_v4i)g,          (lptr_v4i)l,          0, 0);
    __builtin_amdgcn_global_load_async_to_lds_b128(
        (gptr_v4i)(g + 4096), (lptr_v4i)(l + 4096), 0, 0);
}

__global__ __launch_bounds__(256)
void ctc_loss_kernel(const int* __restrict__ labels,   // [B, L] int32
                     const float* __restrict__ ypred,  // [B, T, C] fp32
                     float* __restrict__ out)          // [B, 1] fp32
{
    __shared__ __align__(16) float ybuf[2][TT * C_];  // 2 x 8 KB slabs
    __shared__ float logz[TT];                        // per-t log denominator
    __shared__ float alpha[S_ + 3];                   // CTC forward state
    __shared__ int   lab[L_];

    const int b   = blockIdx.x;
    const int tid = threadIdx.x;
    const float* __restrict__ yrow = ypred + (size_t)b * T_ * C_;

    if (tid < L_) lab[tid] = labels[b * L_ + tid];

    // --- async-stage tile 0 into buffer 0 while we set up ---
    async_stage_tile(yrow, &ybuf[0][0], tid);
    __builtin_amdgcn_s_wait_asynccnt(0);
    __syncthreads();

    // Extended label symbol + skip-transition permission for my state.
    int  ext    = BLANK_;
    bool allow2 = false;
    if (tid < S_ && (tid & 1)) {
        ext    = lab[tid >> 1];
        allow2 = (tid < 3) || (ext != lab[(tid >> 1) - 1]);  // s==1 gated by s>=2 below
    }

    for (int tile = 0; tile < NTILE; ++tile) {
        const int buf = tile & 1;
        float* __restrict__ yb = &ybuf[buf][0];

        // ---- kick off async staging of the NEXT tile into the other buffer ----
        if (tile + 1 < NTILE)
            async_stage_tile(yrow + (size_t)(tile + 1) * TT * C_, &ybuf[buf ^ 1][0], tid);
        // Warm L2 for tile+2 (global_prefetch_b8).
        if (tile + 2 < NTILE)
            __builtin_prefetch(yrow + (size_t)(tile + 2) * TT * C_ + tid * 8, 0, 0);

        // ---- 16 channel-sums via fp32 WMMA (wave 0 only; EXEC all-ones there) ----
        if (tid < 32) {
            const int lane = tid;
            const int row  = lane & 15;   // M  = time row within tile
            const int kq   = lane >> 4;   // lanes 0-15: K=0,1 ; lanes 16-31: K=2,3
            v8f acc = {};
            v2f ones; ones.x = 1.0f; ones.y = 1.0f;
            #pragma unroll
            for (int kk = 0; kk < 32; ++kk) {          // K = 128 in chunks of 4
                v2f a;
                a.x = yb[row * C_ + kk * 4 + 2 * kq + 0];
                a.y = yb[row * C_ + kk * 4 + 2 * kq + 1];
                acc = __builtin_amdgcn_wmma_f32_16x16x4_f32(
                        false, a, false, ones, (short)0, acc, false, false);
            }
            // D layout: lane 0 holds (M=j, N=0) in acc[j]; lane 16 holds M=j+8.
            if (row == 0) {
                #pragma unroll
                for (int j = 0; j < 8; ++j) logz[kq * 8 + j] = acc[j];
            }
        }
        __syncthreads();
        if (tid < TT) logz[tid] = __logf(logz[tid] + (float)C_ * EPS_);
        __syncthreads();

        // ---- sequential CTC scan over this tile's 16 time steps ----
        for (int tl = 0; tl < TT; ++tl) {
            const int t = tile * TT + tl;

            float lp = 0.0f;
            if (tid < S_)
                lp = __logf(yb[tl * C_ + ext] + EPS_) - logz[tl];

            if (t == 0) {                    // uniform branch across the block
                if (tid < S_) alpha[tid] = (tid < 2) ? lp : NEG_;
                __syncthreads();
                continue;
            }

            float a0 = NEG_, a1 = NEG_, a2 = NEG_;
            if (tid < S_) {
                a0 = alpha[tid];
                a1 = (tid >= 1) ? alpha[tid - 1] : NEG_;
                a2 = (tid >= 2 && allow2) ? alpha[tid - 2] : NEG_;
            }
            __syncthreads();                 // all reads of old alpha done
            if (tid < S_) {
                float m   = fmaxf(a0, fmaxf(a1, a2));
                float lse = m + __logf(__expf(a0 - m) + __expf(a1 - m) + __expf(a2 - m));
                alpha[tid] = lp + lse;
            }
            __syncthreads();                 // new alpha visible
        }

        // Next tile's slab must be fully landed in LDS before anyone reads it.
        __builtin_amdgcn_s_wait_asynccnt(0);
        __syncthreads();
    }

    if (tid == 0) {
        const float x = alpha[S_ - 1];
        const float y = alpha[S_ - 2];
        const float m = fmaxf(x, y);
        out[b] = -(m + __logf(__expf(x - m) + __expf(y - m)));
    }
}

extern "C" void kernel_launch(void* const* d_in, const int* in_sizes, int n_in,
                              void* d_out, int out_size, void* d_ws, size_t ws_size,
                              hipStream_t stream) {
    const int*   y_true = (const int*)d_in[0];    // [B, L] (int32 per harness convention)
    const float* y_pred = (const float*)d_in[1];  // [B, T, C] fp32
    float*       out    = (float*)d_out;          // [B, 1] fp32

    ctc_loss_kernel<<<B_, 256, 0, stream>>>(y_true, y_pred, out);
}